// Convert2ImageLayer_48447231099243
// MI455X (gfx1250) — compile-verified
//
#include <hip/hip_runtime.h>
#include <stdint.h>

typedef __attribute__((ext_vector_type(4))) int v4i;

// Problem constants (match reference shapes): B=8, N=1024, C=128, H=W=512
constexpr int Nc = 1024;
constexpr int Cc = 128;
constexpr int Hc = 512;
constexpr int Wc = 512;

constexpr int WAVE            = 32;   // gfx1250 is wave32
constexpr int WAVES_PER_BLOCK = 8;
constexpr int THREADS         = WAVES_PER_BLOCK * WAVE;
constexpr int PIX_PER_WAVE    = 32;   // one 512B feature row per wave per step

// CDNA5 async global->LDS copy path (probe-confirmed to lower on this
// toolchain); plain-load fallback keeps compilation safe elsewhere.
#if defined(__AMDGCN__) &&                                            \
    __has_builtin(__builtin_amdgcn_global_load_async_to_lds_b32) &&   \
    __has_builtin(__builtin_amdgcn_s_wait_asynccnt)
#define USE_ASYNC_IDX 1
#else
#define USE_ASYNC_IDX 0
#endif

#if USE_ASYNC_IDX
typedef __attribute__((address_space(1))) int as1_int;  // global
typedef __attribute__((address_space(3))) int as3_int;  // LDS
#endif

__global__ __launch_bounds__(THREADS) void convert2image_gather(
    const float* __restrict__ feat,   // [B, N, C] f32
    const int*   __restrict__ slic,   // [B, H, W, 1] int32, 1-indexed labels
    float*       __restrict__ out,    // [B, H, W, C] f32
    long long    totalPix)
{
  const int lane      = threadIdx.x & (WAVE - 1);
  const int waveInBlk = threadIdx.x >> 5;
  const long long waveId  = (long long)blockIdx.x * WAVES_PER_BLOCK + waveInBlk;
  const long long pixBase = waveId * PIX_PER_WAVE;  // 32 consecutive pixels
  if (pixBase >= totalPix) return;                  // whole-wave retire only

  int idx;
#if USE_ASYNC_IDX
  __shared__ int sidx[THREADS];
  {
    // Stage this wave's 32 indices into LDS via the gfx1250 async copy path.
    as1_int* gsrc = (as1_int*)(uintptr_t)(slic + pixBase + lane);
    as3_int* ldst = (as3_int*)(unsigned)(uintptr_t)&sidx[threadIdx.x];
    __builtin_amdgcn_global_load_async_to_lds_b32(gsrc, ldst, 0, 0);
    __builtin_amdgcn_s_wait_asynccnt(0);
    idx = sidx[threadIdx.x] - 1;     // wave-private; no barrier needed
  }
#else
  idx = __builtin_nontemporal_load(slic + pixBase + lane) - 1;
#endif

  // All 32 pixels of this wave lie in one batch (H*W divisible by 32).
  const int b = (int)(pixBase / (long long)(Hc * Wc));
  const int* fb = (const int*)(feat + (size_t)b * ((size_t)Nc * Cc)) + (size_t)lane * 4;
  int*       ob = (int*)(out + (size_t)pixBase * Cc) + (size_t)lane * 4;

#pragma unroll
  for (int j = 0; j < PIX_PER_WAVE; ++j) {
    // Broadcast pixel j's segment id into an SGPR -> scalar gather base.
    const int id = __builtin_amdgcn_readlane(idx, j);
    // Branchless validity: scalar-select a safe row, then bit-AND with a
    // uniform all-ones/zero mask (exact zeros, no NaN leakage, no branches
    // fragmenting the VMEM clauses).
    const bool valid = (unsigned)id < (unsigned)Nc;
    const int  safe  = valid ? id : 0;   // s_cselect
    const int  mask  = valid ? -1 : 0;   // s_cselect
    // One wave32 x b128 = exactly one 512B feature row (L2-resident table).
    v4i w = *(const v4i*)(fb + (size_t)safe * Cc);
    w.x &= mask; w.y &= mask; w.z &= mask; w.w &= mask;
    // 1.07 GB streaming output: non-temporal so it doesn't thrash L2.
    __builtin_nontemporal_store(w, (v4i*)(ob + (size_t)j * Cc));
  }
}

extern "C" void kernel_launch(void* const* d_in, const int* in_sizes, int n_in,
                              void* d_out, int out_size, void* d_ws, size_t ws_size,
                              hipStream_t stream) {
  const float* feat = (const float*)d_in[0];  // [B,N,C] f32
  const int*   slic = (const int*)d_in[1];    // [B,H,W,1] int32
  float* out = (float*)d_out;                 // [B,H,W,C] f32

  const long long totalPix = (long long)in_sizes[1];            // B*H*W
  const long long waves    = (totalPix + PIX_PER_WAVE - 1) / PIX_PER_WAVE;
  const int blocks = (int)((waves + WAVES_PER_BLOCK - 1) / WAVES_PER_BLOCK);

  convert2image_gather<<<blocks, THREADS, 0, stream>>>(feat, slic, out, totalPix);
}